// BoxBlur_32684701122563
// MI455X (gfx1250) — compile-verified
//
#include <hip/hip_runtime.h>

// BoxBlur 7x7, reflect padding, f32, on 96 x 1024x1024 images.
// Horizontal pass via V_WMMA_F32_16X16X4_F32 (exact f32 matrix pipe),
// vertical pass via register suffix/prefix sums + cross-half ds_bpermute.
// LDS staging via GLOBAL_LOAD_ASYNC_TO_LDS_B32 (ASYNCcnt) when available.

typedef __attribute__((ext_vector_type(2))) float v2f;
typedef __attribute__((ext_vector_type(8))) float v8f;

#define TILE 16
#define PAD 3
#define PATCH 22          // TILE + 2*PAD
#define LDS_STRIDE 25     // odd stride -> conflict-free on 64 banks
#define WAVES_PER_BLOCK 8

#if __has_builtin(__builtin_amdgcn_global_load_async_to_lds_b32) && \
    __has_builtin(__builtin_amdgcn_s_wait_asynccnt)
#define USE_ASYNC_LDS 1
#else
#define USE_ASYNC_LDS 0
#endif

__device__ __forceinline__ int reflect1024(int i) {
  i = (i < 0) ? -i : i;           // numpy 'reflect': x[-1] -> x[1]
  return (i > 1023) ? (2046 - i) : i;
}

// Swap values between lane halves (lane <-> lane^16) on wave32.
__device__ __forceinline__ float swap_half16(float x) {
  int lane = (int)(threadIdx.x & 31u);
  int idx = ((lane ^ 16) << 2);   // byte index for ds_bpermute
  return __int_as_float(__builtin_amdgcn_ds_bpermute(idx, __float_as_int(x)));
}

__global__ void __launch_bounds__(256)
boxblur7_wmma(const float* __restrict__ src, float* __restrict__ dst, int nimg) {
  __shared__ float lds[WAVES_PER_BLOCK][PATCH * LDS_STRIDE];

  const int lane = (int)(threadIdx.x & 31u);
  const int wave = (int)(threadIdx.x >> 5u);
  const int gtile = blockIdx.x * WAVES_PER_BLOCK + wave;
  const int img = gtile >> 12;          // 4096 tiles per 1024x1024 image
  const int tr = (gtile >> 6) & 63;
  const int tc = gtile & 63;
  const int row0 = tr * TILE;
  const int col0 = tc * TILE;
  const size_t ibase = (size_t)img * (size_t)(1024 * 1024);

  float* ldsw = lds[wave];

  if (img < nimg) {
    // ---- stage 22x22 reflect-padded patch into this wave's LDS region ----
    for (int e = lane; e < PATCH * PATCH; e += 32) {
      int r = e / PATCH;
      int c = e - r * PATCH;
      int gr = reflect1024(row0 + r - PAD);
      int gc = reflect1024(col0 + c - PAD);
      const float* gp = src + (ibase + (size_t)gr * 1024 + (size_t)gc);
      float* lp = ldsw + (r * LDS_STRIDE + c);
#if USE_ASYNC_LDS
      // Direct memory->LDS DMA, per-lane gather addresses on both sides.
      // Toolchain declares generic int* parameters (per diagnostic).
      __builtin_amdgcn_global_load_async_to_lds_b32(
          (int*)gp, (int*)lp, /*offset=*/0, /*cpol=*/0);
#else
      *lp = *gp;
#endif
    }
    if (lane < PATCH) {  // zero K-pad columns 22..23 so garbage never enters WMMA
      ldsw[lane * LDS_STRIDE + 22] = 0.f;
      ldsw[lane * LDS_STRIDE + 23] = 0.f;
    }
#if USE_ASYNC_LDS
    __builtin_amdgcn_s_wait_asynccnt(0);
#endif
  }
  __syncthreads();
  if (img >= nimg) return;  // wave-uniform; active waves keep EXEC all-ones

  const int hi = lane >> 4;   // which 16-lane half
  const int lm = lane & 15;   // position within half (= matrix row / col)

  // A-matrix rows: H0 covers patch rows 0..15, H1 covers rows 16..21 (clamped;
  // clamped rows produce unused H1 rows 6..15).
  const int rA0 = lm;
  int rA1 = 16 + lm;
  if (rA1 > 21) rA1 = 21;

  // Horizontal pass: H[m][n] = sum_{j=n..n+6} Xp[m][j], as Xp(rows x 24) * Bh(24x16)
  // with banded 0/1 Bh synthesized per-lane. K split into 6 chunks of 4.
  v8f h0 = {};
  v8f h1 = {};
#pragma unroll
  for (int t = 0; t < 6; ++t) {
    const int kb = 4 * t + 2 * hi;  // A/B K index held by reg0 in this lane half
    v2f a0, a1, b;
    a0.x = ldsw[rA0 * LDS_STRIDE + kb];
    a0.y = ldsw[rA0 * LDS_STRIDE + kb + 1];
    a1.x = ldsw[rA1 * LDS_STRIDE + kb];
    a1.y = ldsw[rA1 * LDS_STRIDE + kb + 1];
    b.x = (kb >= lm && kb <= lm + 6) ? 1.f : 0.f;
    b.y = (kb + 1 >= lm && kb + 1 <= lm + 6) ? 1.f : 0.f;
    h0 = __builtin_amdgcn_wmma_f32_16x16x4_f32(false, a0, false, b, (short)0,
                                               h0, false, false);
    h1 = __builtin_amdgcn_wmma_f32_16x16x4_f32(false, a1, false, b, (short)0,
                                               h1, false, false);
  }

  // Vertical pass over C/D layout: reg r, lower half = H row r, upper = row r+8.
  // out row R needs H rows R..R+6; rows from the opposite half arrive via swap.
  float S0[6], S1[6], Q[6];
#pragma unroll
  for (int r = 0; r < 6; ++r) {
    S0[r] = swap_half16(h0[r]);  // lower half receives H rows 8..13
    S1[r] = swap_half16(h1[r]);  // upper half receives H rows 16..21
  }
#pragma unroll
  for (int r = 0; r < 6; ++r) Q[r] = hi ? S1[r] : S0[r];

  float QS[6];  // prefix sums of cross-half rows
  QS[0] = Q[0];
#pragma unroll
  for (int r = 1; r < 6; ++r) QS[r] = QS[r - 1] + Q[r];

  float P[8];  // suffix sums of own-half rows: P[r] = sum_{j=r..7} h0[j]
  P[7] = h0[7];
#pragma unroll
  for (int r = 6; r >= 0; --r) P[r] = P[r + 1] + h0[r];

  const float scale = 1.f / 49.f;
#pragma unroll
  for (int r = 0; r < 8; ++r) {
    // out reg r (row R = r + 8*hi): rows R..R+6 =
    //   r==0: h0[0..6]             = P[0] - h0[7]
    //   r==1: h0[1..7]             = P[1]
    //   r>=2: h0[r..7] + Q[0..r-2] = P[r] + QS[r-2]
    float v = P[r];
    if (r == 0) v -= h0[7];
    if (r >= 2) v += QS[r - 2];
    const int orow = row0 + r + 8 * hi;
    // Output is write-once / never re-read: non-temporal store keeps L2 free
    // for input halo reuse.
    __builtin_nontemporal_store(
        v * scale, dst + (ibase + (size_t)orow * 1024 + (size_t)(col0 + lm)));
  }
}

extern "C" void kernel_launch(void* const* d_in, const int* in_sizes, int n_in,
                              void* d_out, int out_size, void* d_ws, size_t ws_size,
                              hipStream_t stream) {
  (void)n_in; (void)out_size; (void)d_ws; (void)ws_size;
  const float* x = (const float*)d_in[0];
  // d_in[1] is the all-ones 7x7 kernel; reference normalizes it, so the
  // normalized weight is the constant 1/49 folded into the kernel above.
  float* out = (float*)d_out;
  const int nimg = in_sizes[0] / (1024 * 1024);     // 96 = 32*3
  const int tiles = nimg * 64 * 64;                 // 16x16 output tiles
  dim3 grid(tiles / WAVES_PER_BLOCK);
  boxblur7_wmma<<<grid, 256, 0, stream>>>(x, out, nimg);
}